// CANLayer_69080253989010
// MI455X (gfx1250) — compile-verified
//
#include <hip/hip_runtime.h>

#define NCELLS 50000
#define NEDGES 800000
#define EPS_SKIP (1.0f + 1e-6f)

typedef float v2f __attribute__((ext_vector_type(2)));
typedef float v8f __attribute__((ext_vector_type(8)));

// ---------------------------------------------------------------------------
// GEMM: C[M,128] = scale * (A[M,128] @ B[128,128] + bias)   (f32 WMMA 16x16x4)
// One wave = one 16x16 output tile. 8 waves per block cover N=128.
// A tile staged in LDS (padded stride 132 -> conflict-free ds reads).
// ---------------------------------------------------------------------------
__global__ __launch_bounds__(256) void gemm128_wmma(
    const float* __restrict__ A, const float* __restrict__ B,
    float* __restrict__ C, const float* __restrict__ bias, float scale)
{
    __shared__ float Atile[16 * 132];
    const int tid   = threadIdx.x;
    const int mtile = blockIdx.x;

    // Stage 16x128 A tile: 512 float4 chunks, 2 per thread, coalesced.
    {
        const float4* A4 = (const float4*)(A + (size_t)mtile * 16 * 128);
#pragma unroll
        for (int i = 0; i < 2; ++i) {
            int idx = tid + i * 256;      // 0..511
            int r   = idx >> 5;           // row 0..15
            int c4  = idx & 31;           // float4 column
            float4 v = A4[r * 32 + c4];
            *(float4*)(&Atile[r * 132 + c4 * 4]) = v;
        }
    }
    __syncthreads();

    const int lane = tid & 31;
    const int wv   = tid >> 5;            // N-tile 0..7
    const int half = lane >> 4;           // 0: K pair {k,k+1}; 1: {k+2,k+3}
    const int m    = lane & 15;           // A row / B column within tile
    const int ncol = (wv << 4) + m;       // global N column

    v8f c = {};
#pragma unroll
    for (int kk = 0; kk < 128; kk += 4) {
        const int k0 = kk + 2 * half;
        v2f a, b;
        a.x = Atile[m * 132 + k0];
        a.y = Atile[m * 132 + k0 + 1];
        b.x = B[k0 * 128 + ncol];
        b.y = B[(k0 + 1) * 128 + ncol];
        c = __builtin_amdgcn_wmma_f32_16x16x4_f32(false, a, false, b,
                                                  (short)0, c, false, false);
    }

    const float bv = bias ? bias[ncol] : 0.0f;
#pragma unroll
    for (int r = 0; r < 8; ++r) {
        int row = mtile * 16 + r + 8 * half;   // C: vgpr r -> M=r / M=r+8
        C[(size_t)row * 128 + ncol] = scale * (c[r] + bv);
    }
}

// ---------------------------------------------------------------------------
// Per-node attention scalars: s_src[n,h] = xm[n,h,:]·a_src[h,:], same for dst.
// ---------------------------------------------------------------------------
__global__ void node_scores(const float* __restrict__ xm,
                            const float* __restrict__ a_src,
                            const float* __restrict__ a_dst,
                            float* __restrict__ s_src,
                            float* __restrict__ s_dst)
{
    int t = blockIdx.x * blockDim.x + threadIdx.x;
    if (t >= NCELLS * 4) return;
    int n = t >> 2, h = t & 3;
    const float4* xr = (const float4*)(xm + (size_t)n * 128 + h * 32);
    const float4* as = (const float4*)(a_src + h * 32);
    const float4* ad = (const float4*)(a_dst + h * 32);
    float ss = 0.f, sd = 0.f;
#pragma unroll
    for (int i = 0; i < 8; ++i) {
        float4 v = xr[i], p = as[i], q = ad[i];
        ss += v.x * p.x + v.y * p.y + v.z * p.z + v.w * p.w;
        sd += v.x * q.x + v.y * q.y + v.z * q.z + v.w * q.w;
    }
    s_src[t] = ss;
    s_dst[t] = sd;
}

__global__ void init_buffers(float* __restrict__ denoms, int ndenom,
                             float* __restrict__ maxes, int nmax)
{
    int t = blockIdx.x * blockDim.x + threadIdx.x;
    int stride = gridDim.x * blockDim.x;
    for (int i = t; i < ndenom; i += stride) denoms[i] = 0.f;
    if (t < nmax) maxes[t] = -__builtin_inff();
}

__device__ inline float lrelu(float z) { return z >= 0.f ? z : 0.01f * z; }

// Sign-aware float atomic max via int/uint ordering trick (init = -inf).
__device__ inline void atomicMaxF(float* addr, float v) {
    if (v >= 0.f) atomicMax((int*)addr, __float_as_int(v));
    else          atomicMin((unsigned int*)addr, __float_as_uint(v));
}

// ---------------------------------------------------------------------------
// Pass 1 over edges: global per-head max of leaky_relu(alpha).
// ---------------------------------------------------------------------------
__global__ __launch_bounds__(256) void edge_alpha_max(
    const int* __restrict__ tgt, const int* __restrict__ src,
    const float* __restrict__ s_src, const float* __restrict__ s_dst,
    float* __restrict__ gmax)
{
    __shared__ float sm[4];
    if (threadIdx.x < 4) sm[threadIdx.x] = -__builtin_inff();
    __syncthreads();
    float lm0 = -__builtin_inff(), lm1 = lm0, lm2 = lm0, lm3 = lm0;
    int stride = gridDim.x * blockDim.x;
    for (int e = blockIdx.x * blockDim.x + threadIdx.x; e < NEDGES; e += stride) {
        int s = src[e] << 2, t = tgt[e] << 2;
        lm0 = fmaxf(lm0, lrelu(s_src[s + 0] + s_dst[t + 0]));
        lm1 = fmaxf(lm1, lrelu(s_src[s + 1] + s_dst[t + 1]));
        lm2 = fmaxf(lm2, lrelu(s_src[s + 2] + s_dst[t + 2]));
        lm3 = fmaxf(lm3, lrelu(s_src[s + 3] + s_dst[t + 3]));
    }
    atomicMaxF(&sm[0], lm0);
    atomicMaxF(&sm[1], lm1);
    atomicMaxF(&sm[2], lm2);
    atomicMaxF(&sm[3], lm3);
    __syncthreads();
    if (threadIdx.x < 4) atomicMaxF(&gmax[threadIdx.x], sm[threadIdx.x]);
}

// ---------------------------------------------------------------------------
// Pass 2 over edges: denom[tgt,h] += exp(alpha - gmax[h]).
// ---------------------------------------------------------------------------
__global__ void edge_exp_denom(
    const int* __restrict__ tgt, const int* __restrict__ src,
    const float* __restrict__ s_src, const float* __restrict__ s_dst,
    const float* __restrict__ gmax, float* __restrict__ denom)
{
    int e = blockIdx.x * blockDim.x + threadIdx.x;
    if (e >= NEDGES) return;
    int s = src[e] << 2, t = tgt[e] << 2;
#pragma unroll
    for (int h = 0; h < 4; ++h) {
        float z = lrelu(s_src[s + h] + s_dst[t + h]);
        atomicAdd(&denom[t + h], __expf(z - gmax[h]));
    }
}

// ---------------------------------------------------------------------------
// Pass 3 over edges (heavy): out[tgt] += xm[src] * a.  One wave32 per edge;
// lane covers 4 contiguous floats (head = lane>>3).  float4 coalesced gather
// + 4x global_atomic_add_f32 scatter.
// ---------------------------------------------------------------------------
__global__ __launch_bounds__(256) void edge_aggregate(
    const int* __restrict__ tgt, const int* __restrict__ src,
    const float* __restrict__ s_src, const float* __restrict__ s_dst,
    const float* __restrict__ gmax, const float* __restrict__ denom,
    const float* __restrict__ xm, float* __restrict__ out)
{
    const int lane   = threadIdx.x & 31;
    const int wave   = (blockIdx.x * blockDim.x + threadIdx.x) >> 5;
    const int nwaves = (gridDim.x * blockDim.x) >> 5;
    const int h      = lane >> 3;
    for (int e = wave; e < NEDGES; e += nwaves) {
        int s = src[e], t = tgt[e];
        float z  = lrelu(s_src[s * 4 + h] + s_dst[t * 4 + h]);
        float ev = __expf(z - gmax[h]);
        float a  = ev / (denom[t * 4 + h] + 1e-16f);
        float4 xs = *(const float4*)(xm + (size_t)s * 128 + lane * 4);
        float* op = out + (size_t)t * 128 + lane * 4;
        atomicAdd(op + 0, xs.x * a);
        atomicAdd(op + 1, xs.y * a);
        atomicAdd(op + 2, xs.z * a);
        atomicAdd(op + 3, xs.w * a);
    }
}

__global__ void relu_inplace(float* __restrict__ out, int n)
{
    int i = blockIdx.x * blockDim.x + threadIdx.x;
    if (i < n) out[i] = fmaxf(out[i], 0.f);
}

// ---------------------------------------------------------------------------
extern "C" void kernel_launch(void* const* d_in, const int* in_sizes, int n_in,
                              void* d_out, int out_size, void* d_ws, size_t ws_size,
                              hipStream_t stream)
{
    const float* x         = (const float*)d_in[0];
    const int*   ltgt      = (const int*)  d_in[1];
    const int*   lsrc      = (const int*)  d_in[2];
    const int*   utgt      = (const int*)  d_in[3];
    const int*   usrc      = (const int*)  d_in[4];
    const float* W_low     = (const float*)d_in[5];
    const float* a_src_low = (const float*)d_in[6];
    const float* a_dst_low = (const float*)d_in[7];
    const float* W_up      = (const float*)d_in[8];
    const float* a_src_up  = (const float*)d_in[9];
    const float* a_dst_up  = (const float*)d_in[10];
    const float* W_skip    = (const float*)d_in[11];
    const float* b_skip    = (const float*)d_in[12];
    float* out = (float*)d_out;

    float* ws        = (float*)d_ws;
    float* xm_low    = ws;  ws += (size_t)NCELLS * 128;
    float* xm_up     = ws;  ws += (size_t)NCELLS * 128;
    float* s_src_low = ws;  ws += NCELLS * 4;
    float* s_dst_low = ws;  ws += NCELLS * 4;
    float* s_src_up  = ws;  ws += NCELLS * 4;
    float* s_dst_up  = ws;  ws += NCELLS * 4;
    float* denom     = ws;  ws += NCELLS * 8;   // [low | up]
    float* gmax      = ws;  ws += 8;            // [low 4 | up 4]
    float* denom_up  = denom + NCELLS * 4;

    const int mtiles = NCELLS / 16;             // 50000 = 3125*16 exactly

    // GEMMs (skip path written directly into out, pre-scaled by EPS)
    gemm128_wmma<<<mtiles, 256, 0, stream>>>(x, W_skip, out, b_skip, EPS_SKIP);
    gemm128_wmma<<<mtiles, 256, 0, stream>>>(x, W_low, xm_low, nullptr, 1.0f);
    gemm128_wmma<<<mtiles, 256, 0, stream>>>(x, W_up,  xm_up,  nullptr, 1.0f);

    init_buffers<<<1024, 256, 0, stream>>>(denom, NCELLS * 8, gmax, 8);

    const int nsBlocks = (NCELLS * 4 + 255) / 256;
    node_scores<<<nsBlocks, 256, 0, stream>>>(xm_low, a_src_low, a_dst_low,
                                              s_src_low, s_dst_low);
    node_scores<<<nsBlocks, 256, 0, stream>>>(xm_up, a_src_up, a_dst_up,
                                              s_src_up, s_dst_up);

    edge_alpha_max<<<512, 256, 0, stream>>>(ltgt, lsrc, s_src_low, s_dst_low, gmax);
    edge_alpha_max<<<512, 256, 0, stream>>>(utgt, usrc, s_src_up,  s_dst_up,  gmax + 4);

    const int edBlocks = (NEDGES + 255) / 256;
    edge_exp_denom<<<edBlocks, 256, 0, stream>>>(ltgt, lsrc, s_src_low, s_dst_low,
                                                 gmax, denom);
    edge_exp_denom<<<edBlocks, 256, 0, stream>>>(utgt, usrc, s_src_up, s_dst_up,
                                                 gmax + 4, denom_up);

    edge_aggregate<<<4096, 256, 0, stream>>>(ltgt, lsrc, s_src_low, s_dst_low,
                                             gmax, denom, xm_low, out);
    edge_aggregate<<<4096, 256, 0, stream>>>(utgt, usrc, s_src_up, s_dst_up,
                                             gmax + 4, denom_up, xm_up, out);

    relu_inplace<<<(NCELLS * 128 + 255) / 256, 256, 0, stream>>>(out, NCELLS * 128);
}